// CtcLoss_36386962931718
// MI455X (gfx1250) — compile-verified
//
#include <hip/hip_runtime.h>
#include <stdint.h>

#define NEGV (-1e30f)

static constexpr int B  = 32;
static constexpr int T  = 1000;
static constexpr int V  = 1024;
static constexpr int L  = 100;
static constexpr int S  = 2 * L + 1;   // 201 extended states
static constexpr int SP = 104;         // padded gathered row, floats (416 B = 26 * 16 B)
static constexpr int NLP = SP / 4;     // 26 async b128 loads per row
static constexpr int WPB = 8;          // waves per block in kernel 1

// ---------------- CDNA5 async global->LDS helpers ----------------
__device__ __forceinline__ uint32_t lds_off(const void* p) {
  // generic pointer to LDS: low 32 bits are the LDS byte address
  return (uint32_t)(uintptr_t)p;
}

__device__ __forceinline__ void async_load_b128(uint32_t loff, const void* gaddr) {
  // GLOBAL_LOAD_ASYNC_TO_LDS_B128: VDST = LDS byte address, VADDR = 64-bit global addr
  asm volatile("global_load_async_to_lds_b128 %0, %1, off"
               :: "v"(loff), "v"(gaddr)
               : "memory");
}

__device__ __forceinline__ void wait_async0() {
#if __has_builtin(__builtin_amdgcn_s_wait_asynccnt)
  __builtin_amdgcn_s_wait_asynccnt(0);
#else
  asm volatile("s_wait_asynccnt 0" ::: "memory");
#endif
}

// ---------------- math ----------------
__device__ __forceinline__ float lae(float a, float b) {
  float mx = fmaxf(a, b);
  float mn = fminf(a, b);
  // mn - mx <= 0; exp in (0,1]; fast transcendentals (v_exp_f32 / v_log_f32)
  return mx + __logf(1.0f + __expf(mn - mx));
}

// ---------------- kernel 1: row logsumexp + symbol gather ----------------
// One wave32 per (b,t) vocab row: 8 outstanding b128 loads/lane, pure in-wave
// butterfly reductions (no LDS, no barriers). grid = B*T/WPB blocks x 256 thr.
__global__ __launch_bounds__(256)
void rowlse_gather(const float* __restrict__ pred,
                   const int* __restrict__ gt,
                   float* __restrict__ lpg) {
  const int wave = threadIdx.x >> 5;
  const int lane = threadIdx.x & 31;
  const int row  = blockIdx.x * WPB + wave;   // b*T + t
  const int b    = row / T;

  const float*  __restrict__ p  = pred + (size_t)row * V;
  const float4* __restrict__ p4 = reinterpret_cast<const float4*>(p);

  float4 v[8];
  #pragma unroll
  for (int k = 0; k < 8; ++k) v[k] = p4[lane + 32 * k];   // 8x global_load_b128

  // lane-local max of 32 values
  float m = fmaxf(fmaxf(v[0].x, v[0].y), fmaxf(v[0].z, v[0].w));
  #pragma unroll
  for (int k = 1; k < 8; ++k) {
    m = fmaxf(m, fmaxf(fmaxf(v[k].x, v[k].y), fmaxf(v[k].z, v[k].w)));
  }
  // wave32 butterfly max: all lanes end with row max
  #pragma unroll
  for (int off = 16; off; off >>= 1) m = fmaxf(m, __shfl_xor(m, off, 32));

  // lane-local sum of exp
  float s = 0.0f;
  #pragma unroll
  for (int k = 0; k < 8; ++k) {
    s += __expf(v[k].x - m) + __expf(v[k].y - m) +
         __expf(v[k].z - m) + __expf(v[k].w - m);
  }
  #pragma unroll
  for (int off = 16; off; off >>= 1) s += __shfl_xor(s, off, 32);

  const float logZ = m + __logf(s);

  // gather log-probs at blank (j=0) and the L labels (j=1..L); row is L0-hot
  float* __restrict__ orow = lpg + (size_t)row * SP;
  #pragma unroll
  for (int jj = lane; jj <= L; jj += 32) {
    const int idx = (jj == 0) ? 0 : gt[b * L + jj - 1];
    orow[jj] = p[idx] - logZ;
  }
}

// ---------------- kernel 2: CTC alpha recursion ----------------
// grid = B blocks, 224 threads (7 waves); one thread per extended state.
// lp rows streamed L2->LDS with async DMA, double-buffered, 1 barrier/step.
__global__ __launch_bounds__(224)
void ctc_scan(const float* __restrict__ lpg,
              const int* __restrict__ gt,
              const int* __restrict__ plen,
              const int* __restrict__ glen,
              float* __restrict__ loss_out) {
  const int b   = blockIdx.x;
  const int tid = threadIdx.x;

  __shared__ float alpha[2][S + 3];
  __shared__ __align__(16) float lp[2][SP];
  __shared__ int gts[L];

  const int Tb = plen[b];
  const int tl = glen[b];
  if (tid < L) gts[tid] = gt[b * L + tid];

  const float* __restrict__ lrow = lpg + (size_t)b * T * SP;

  // prefetch lp rows for t=0 (and t=1) via async DMA into LDS
  if (tid < NLP) {
    async_load_b128(lds_off(&lp[0][tid * 4]), lrow + tid * 4);
    if (Tb > 1)
      async_load_b128(lds_off(&lp[1][tid * 4]), lrow + SP + tid * 4);
  }
  wait_async0();
  __syncthreads();

  // per-state constants (after gts visible)
  int j = 0, skip = 0;
  if (tid < S) {
    if (tid & 1) {
      const int k = tid >> 1;           // 0..L-1
      j = 1 + k;
      skip = (tid >= 3) && (gts[k] != gts[k - 1]);
    }
  }

  // t = 0 init
  if (tid < S) {
    float a0 = NEGV;
    if (tid == 0)                 a0 = lp[0][0];
    else if (tid == 1 && tl > 0)  a0 = lp[0][1];
    alpha[0][tid] = a0;
  }
  __syncthreads();

  for (int t = 1; t < Tb; ++t) {
    const int cur = (t - 1) & 1;
    const int nxt = t & 1;
    // prefetch lp for t+1 while computing step t
    if (tid < NLP && (t + 1) < Tb)
      async_load_b128(lds_off(&lp[(t + 1) & 1][tid * 4]),
                      lrow + (size_t)(t + 1) * SP + tid * 4);
    if (tid < S) {
      const float ac = alpha[cur][tid];
      const float p1 = (tid >= 1) ? alpha[cur][tid - 1] : NEGV;
      float a = lae(ac, p1);
      if (skip) a = lae(a, alpha[cur][tid - 2]);
      alpha[nxt][tid] = a + lp[t & 1][j];
    }
    wait_async0();        // issuing wave drains ASYNCcnt; others no-op
    __syncthreads();      // alpha + lp double-buffer handoff
  }

  if (tid == 0) {
    const int fin = (Tb - 1) & 1;       // Tb==1 -> alpha0
    const float e1 = alpha[fin][2 * tl];
    int i2 = 2 * tl - 1; if (i2 < 0) i2 = 0;
    const float e2 = alpha[fin][i2];
    const float f  = lae(e1, e2);
    const float l  = (f <= NEGV * 0.5f) ? 0.0f : -f;   // zero_infinity
    loss_out[b] = l / (float)tl;
  }
}

// ---------------- kernel 3: deterministic mean over batch ----------------
__global__ __launch_bounds__(32)
void reduce_mean(const float* __restrict__ loss, float* __restrict__ out, int n) {
  float v = (threadIdx.x < n) ? loss[threadIdx.x] : 0.0f;
  #pragma unroll
  for (int off = 16; off; off >>= 1) v += __shfl_xor(v, off, 32);
  if (threadIdx.x == 0) out[0] = v / (float)n;
}

extern "C" void kernel_launch(void* const* d_in, const int* in_sizes, int n_in,
                              void* d_out, int out_size, void* d_ws, size_t ws_size,
                              hipStream_t stream) {
  const float* pred = (const float*)d_in[0];   // [B,T,V] fp32
  const int*   plen = (const int*)d_in[1];     // [B]
  const int*   gt   = (const int*)d_in[2];     // [B,L]
  const int*   glen = (const int*)d_in[3];     // [B]

  float* lpg   = (float*)d_ws;                                   // [B,T,SP]
  float* lossb = (float*)((char*)d_ws + (size_t)B * T * SP * sizeof(float));

  rowlse_gather<<<(B * T) / WPB, 32 * WPB, 0, stream>>>(pred, gt, lpg);
  ctc_scan<<<B, 224, 0, stream>>>(lpg, gt, plen, glen, lossb);
  reduce_mean<<<1, 32, 0, stream>>>(lossb, (float*)d_out, B);
}